// QCONV2d_5x5_35880156791524
// MI455X (gfx1250) — compile-verified
//
#include <hip/hip_runtime.h>

#define KSIZE 5
#define HW    64
#define PLANE (HW * HW)          // 4096 elements per (b,c) plane
#define AW    68                 // 64 + 2*2 halo
#define NTAP  25
#define TPB   256
#define PIX_PER_THREAD (PLANE / TPB)   // 16

typedef unsigned int u32x4 __attribute__((ext_vector_type(4)));
typedef int          i32x8 __attribute__((ext_vector_type(8)));
typedef int          i32x4 __attribute__((ext_vector_type(4)));

__global__ __launch_bounds__(TPB)
void quanv5x5_kernel(const float* __restrict__ x,
                     const float* __restrict__ w,
                     float* __restrict__ out) {
    __shared__ float s_x[PLANE];        // raw plane staged by TDM
    __shared__ float s_c[AW * AW];      // cos(x) with (1,0) halo
    __shared__ float s_s[AW * AW];      // sin(x) with (1,0) halo
    __shared__ float s_cw[NTAP];
    __shared__ float s_sw[NTAP];

    const int tid   = threadIdx.x;
    const int plane = blockIdx.x;
    const float* xp = x + (size_t)plane * PLANE;

    // ---- Wave 0: issue one Tensor-Data-Mover load of the whole plane ----
    if (tid < 32) {
        unsigned long long gaddr = (unsigned long long)(uintptr_t)xp;
        unsigned int lds_off = (unsigned int)(uintptr_t)(&s_x[0]); // low 32b = LDS byte offset

        u32x4 g0;
        g0[0] = 1u;                                   // count=1, user mode, no gather
        g0[1] = lds_off;                              // lds_addr
        g0[2] = (unsigned int)gaddr;                  // global_addr[31:0]
        g0[3] = ((unsigned int)(gaddr >> 32) & 0x01FFFFFFu) | (2u << 30); // addr[56:32] | type=2

        i32x8 g1;
        g1[0] = 0x00020000;                           // wg_mask=0, data_size=2 (4 bytes)
        g1[1] = (int)(((unsigned)PLANE & 0xFFFFu) << 16);            // tensor_dim0 lo16 @bits63:48
        g1[2] = (int)((((unsigned)PLANE >> 16) & 0xFFFFu) | (1u << 16)); // dim0 hi | tensor_dim1=1 lo
        g1[3] = (int)((unsigned)PLANE << 16);         // tensor_dim1 hi=0 | tile_dim0=4096
        g1[4] = 1;                                    // tile_dim1=1, tile_dim2=0
        g1[5] = PLANE;                                // tensor_dim0_stride lo32
        g1[6] = (int)((unsigned)PLANE << 16);         // stride0 hi=0 | tensor_dim1_stride lo16
        g1[7] = 0;                                    // stride1 hi

        i32x4 gz4 = {0, 0, 0, 0};                     // groups 2/3 unused (<=2D tensor)
        i32x8 gz8 = {0, 0, 0, 0, 0, 0, 0, 0};        // extra group (clang-23 6-arg form)
        __builtin_amdgcn_tensor_load_to_lds(g0, g1, gz4, gz4, gz8, 0);
    }

    // ---- Per-tap constants: cos(w_k), sin(w_k) ----
    if (tid < NTAP) {
        float wv = w[tid];
        s_cw[tid] = __cosf(wv);
        s_sw[tid] = __sinf(wv);
    }

    // ---- Init full haloed arrays to the zero-pad contribution (cos=1, sin=0) ----
    #pragma unroll 1
    for (int i = tid; i < AW * AW; i += TPB) {
        s_c[i] = 1.0f;
        s_s[i] = 0.0f;
    }

    if (tid < 32) __builtin_amdgcn_s_wait_tensorcnt(0);   // TDM data landed in LDS
    __syncthreads();

    // ---- sincos once per input pixel (the 12x trans reduction) ----
    #pragma unroll 1
    for (int i = 0; i < PIX_PER_THREAD; ++i) {
        int p = tid + i * TPB;
        float v = s_x[p];
        int idx = ((p >> 6) + 2) * AW + ((p & 63) + 2);
        s_c[idx] = __cosf(v);
        s_s[idx] = __sinf(v);
    }
    __syncthreads();

    // ---- Hoist tap constants into registers ----
    float cwr[NTAP], swr[NTAP];
    #pragma unroll
    for (int k = 0; k < NTAP; ++k) { cwr[k] = s_cw[k]; swr[k] = s_sw[k]; }

    // ---- 24-tap product per output, coalesced store ----
    float* op = out + (size_t)plane * PLANE;
    #pragma unroll 1
    for (int i = 0; i < PIX_PER_THREAD; ++i) {
        int p  = tid + i * TPB;
        int h  = p >> 6;
        int wv = p & 63;
        float prod = 1.0f;
        #pragma unroll
        for (int di = 0; di < KSIZE; ++di) {
            #pragma unroll
            for (int dj = 0; dj < KSIZE; ++dj) {
                const int k = di * KSIZE + dj;
                if (k == 0) continue;   // patch index 0 excluded from the product
                int a = (h + di) * AW + (wv + dj);
                float cc = s_c[a];
                float ss = s_s[a];
                // cos(x + w_k) = cos(x)cos(w_k) - sin(x)sin(w_k)
                prod *= __builtin_fmaf(cc, cwr[k], -(ss * swr[k]));
            }
        }
        op[p] = prod;
    }
}

extern "C" void kernel_launch(void* const* d_in, const int* in_sizes, int n_in,
                              void* d_out, int out_size, void* d_ws, size_t ws_size,
                              hipStream_t stream) {
    const float* x = (const float*)d_in[0];
    const float* w = (const float*)d_in[1];
    float* out = (float*)d_out;

    int nplanes = in_sizes[0] / PLANE;   // B*C = 512 for the reference shapes
    quanv5x5_kernel<<<dim3(nplanes), dim3(TPB), 0, stream>>>(x, w, out);
}